// CausalGNN_59253368815959
// MI455X (gfx1250) — compile-verified
//
#include <hip/hip_runtime.h>

// ---------------------------------------------------------------------------
// CausalGNN forward for MI455X (gfx1250, wave32).
//   L x ( GCNConv -> BatchNorm(train) -> ReLU[not last] ) with residuals.
// GEMM uses V_WMMA_F32_16X16X4_F32 (exact fp32 matrix cores), with W staged
// in LDS as K-pair-interleaved float2 (one ds_load_b64 per B fragment) and an
// XOR-16 bank swizzle so the two half-waves hit disjoint LDS bank halves.
// Edge aggregation (the bandwidth bottleneck, ~0.9 GB/layer) uses
// float4 gathers + global_atomic_add_f32 scatter.
// ---------------------------------------------------------------------------

typedef float v2f __attribute__((ext_vector_type(2)));
typedef float v8f __attribute__((ext_vector_type(8)));

static constexpr int DFEAT = 128;   // feature dim (fixed by reference)
static constexpr float BN_EPS = 1e-5f;

// ------------------------------ GEMM: H = X @ W -----------------------------
// Block = 256 threads = 8 waves. Block computes 128 rows x 128 cols.
// LDS holds W as Wp[k/2][n] = (W[k][n], W[k+1][n]) with column-XOR swizzle on
// odd k/2 rows. Each wave: 16-row strip, 8 N-tiles, K-loop of 32 steps.
__global__ __launch_bounds__(256)
void gemm_xw_wmma(const float* __restrict__ X, const float* __restrict__ W,
                  float* __restrict__ H, int nrows)
{
    __shared__ v2f Wp[(DFEAT / 2) * DFEAT];   // 64 KB
    const int tid = threadIdx.x;

    {   // cooperative load + interleave: 4096 float4 / 256 threads = 16 each
        const float4* W4 = (const float4*)W;
#pragma unroll
        for (int i = 0; i < 16; ++i) {
            const int idx = tid + 256 * i;         // float4 index
            const float4 v = W4[idx];
            const int k  = idx >> 5;               // source row 0..127
            const int n0 = (idx & 31) << 2;        // source col 0,4,...,124
            const int r2 = k >> 1;                 // pair row 0..63
            const int ns = n0 ^ ((r2 & 1) << 4);   // bank swizzle (same 16-blk)
            float* dstp = (float*)&Wp[r2 * DFEAT + ns];
            const int sub = k & 1;
            dstp[0 * 2 + sub] = v.x;
            dstp[1 * 2 + sub] = v.y;
            dstp[2 * 2 + sub] = v.z;
            dstp[3 * 2 + sub] = v.w;
        }
    }
    __syncthreads();

    const int wave = tid >> 5;
    const int lane = tid & 31;
    const int half = lane >> 4;     // 0: lanes 0-15, 1: lanes 16-31
    const int l15  = lane & 15;
    const int m0   = blockIdx.x * 128 + wave * 16;
    const int m    = m0 + l15;      // A-fragment row for this lane
    // Clamp instead of predicating the load: D row m depends only on A row m,
    // so out-of-range rows may read garbage (store is guarded). Keeps EXEC
    // all-ones (WMMA requirement) and removes per-iteration exec save/restore.
    const int mc   = (m < nrows) ? m : (nrows - 1);
    const float* xrow = X + (size_t)mc * DFEAT;

    v8f acc[8];
#pragma unroll
    for (int t = 0; t < 8; ++t) acc[t] = (v8f)(0.0f);

    for (int k = 0; k < DFEAT; k += 4) {
        // A 16x4 layout: lanes 0-15 -> K = k,k+1 ; lanes 16-31 -> K = k+2,k+3
        v2f a = *(const v2f*)(xrow + k + half * 2);
        const int kk2 = (k >> 1) + half;           // pair row; parity == half
#pragma unroll
        for (int t = 0; t < 8; ++t) {
            // B 4x16: one aligned ds_load_b64; swizzled column = XOR tile id
            v2f b = Wp[kk2 * DFEAT + ((t ^ half) * 16 + l15)];
            acc[t] = __builtin_amdgcn_wmma_f32_16x16x4_f32(
                false, a, false, b, (short)0, acc[t], false, false);
        }
    }

    // C/D layout: VGPR j -> row m0 + j (lanes 0-15) / m0 + 8 + j (lanes 16-31)
#pragma unroll
    for (int t = 0; t < 8; ++t) {
#pragma unroll
        for (int j = 0; j < 8; ++j) {
            const int row = m0 + half * 8 + j;
            if (row < nrows)
                H[(size_t)row * DFEAT + t * 16 + l15] = acc[t][j];
        }
    }
}

// ------------------------- degree / normalization ---------------------------
__global__ __launch_bounds__(256)
void deg_init(float* __restrict__ deg, int n)
{
    int i = blockIdx.x * 256 + threadIdx.x;
    if (i < n) deg[i] = 1.0f;          // self-loop contribution
}

__global__ __launch_bounds__(256)
void deg_count(const int* __restrict__ dst, float* __restrict__ deg, int e)
{
    int i = blockIdx.x * 256 + threadIdx.x;
    if (i < e) atomicAdd(&deg[dst[i]], 1.0f);
}

__global__ __launch_bounds__(256)
void make_dis(float* __restrict__ deg, int n)   // deg -> deg^{-1/2} in place
{
    int i = blockIdx.x * 256 + threadIdx.x;
    if (i < n) deg[i] = rsqrtf(deg[i]);
}

__global__ __launch_bounds__(256)
void make_norm(const int* __restrict__ src, const int* __restrict__ dst,
               const float* __restrict__ dis, float* __restrict__ nrm, int e)
{
    int i = blockIdx.x * 256 + threadIdx.x;
    if (i < e) nrm[i] = dis[src[i]] * dis[dst[i]];
}

// --------------------- agg = (1/deg) * h + bias (self loop) -----------------
__global__ __launch_bounds__(256)
void init_agg(const float* __restrict__ h, const float* __restrict__ dis,
              const float* __restrict__ bias, float* __restrict__ agg, int n)
{
    int i = blockIdx.x * 256 + threadIdx.x;          // float4 element index
    if (i >= n * (DFEAT / 4)) return;
    int node = i >> 5;                               // 32 float4 per node
    int d4   = i & 31;
    float s = dis[node];
    s = s * s;                                       // self_norm = 1/deg
    float4 hv = ((const float4*)h)[i];
    float4 bv = ((const float4*)bias)[d4];
    float4 o;
    o.x = fmaf(s, hv.x, bv.x);
    o.y = fmaf(s, hv.y, bv.y);
    o.z = fmaf(s, hv.z, bv.z);
    o.w = fmaf(s, hv.w, bv.w);
    ((float4*)agg)[i] = o;
}

// ------------------ scatter: agg[dst] += norm[e] * h[src] -------------------
// One wave per edge; lane owns 4 contiguous features (512B per wave).
__global__ __launch_bounds__(256)
void scatter_edges(const int* __restrict__ src, const int* __restrict__ dst,
                   const float* __restrict__ nrm, const float* __restrict__ h,
                   float* __restrict__ agg, int e)
{
    int gid  = blockIdx.x * 256 + threadIdx.x;
    int edge = gid >> 5;
    int lane = gid & 31;
    if (edge >= e) return;
    const int   s = src[edge];
    const int   d = dst[edge];
    const float w = nrm[edge];
    const float4 hv = ((const float4*)(h + (size_t)s * DFEAT))[lane];
    float* ap = agg + (size_t)d * DFEAT + lane * 4;
    atomicAdd(ap + 0, w * hv.x);
    atomicAdd(ap + 1, w * hv.y);
    atomicAdd(ap + 2, w * hv.z);
    atomicAdd(ap + 3, w * hv.w);
}

// ----------------------- BatchNorm statistics (train) -----------------------
__global__ void zero_stats(float* __restrict__ sums, float* __restrict__ sumsq)
{
    int t = threadIdx.x;
    if (t < DFEAT) { sums[t] = 0.0f; sumsq[t] = 0.0f; }
}

// Block handles 512 rows; thread (col = t&127, half = t>>7) strides by 2 rows.
__global__ __launch_bounds__(256)
void col_stats(const float* __restrict__ agg, float* __restrict__ sums,
               float* __restrict__ sumsq, int n)
{
    __shared__ float shs[256];
    __shared__ float shq[256];
    const int col  = threadIdx.x & (DFEAT - 1);
    const int half = threadIdx.x >> 7;
    int r    = blockIdx.x * 512 + half;
    int rend = blockIdx.x * 512 + 512;
    if (rend > n) rend = n;
    float s = 0.0f, q = 0.0f;
    for (; r < rend; r += 2) {
        float v = agg[(size_t)r * DFEAT + col];
        s += v;
        q  = fmaf(v, v, q);
    }
    shs[threadIdx.x] = s;
    shq[threadIdx.x] = q;
    __syncthreads();
    if (half == 0) {
        atomicAdd(&sums[col],  s + shs[threadIdx.x + 128]);
        atomicAdd(&sumsq[col], q + shq[threadIdx.x + 128]);
    }
}

__global__ void bn_finalize(const float* __restrict__ sums,
                            const float* __restrict__ sumsq,
                            const float* __restrict__ gamma,
                            const float* __restrict__ beta,
                            float* __restrict__ scale,
                            float* __restrict__ shift, int n)
{
    int c = threadIdx.x;
    if (c < DFEAT) {
        float inv_n = 1.0f / (float)n;
        float mean  = sums[c] * inv_n;
        float var   = sumsq[c] * inv_n - mean * mean;   // biased batch var
        float sc    = gamma[c] * rsqrtf(var + BN_EPS);
        scale[c] = sc;
        shift[c] = beta[c] - mean * sc;
    }
}

// ------------- y = scale*agg + shift; ReLU?; x_new = x_old + y? -------------
__global__ __launch_bounds__(256)
void bn_apply(const float* __restrict__ agg, const float* __restrict__ scale,
              const float* __restrict__ shift, const float* __restrict__ xin,
              float* __restrict__ xout, int n, int do_relu, int do_resid)
{
    int i = blockIdx.x * 256 + threadIdx.x;
    if (i >= n * (DFEAT / 4)) return;
    int d4 = i & 31;
    float4 a  = ((const float4*)agg)[i];
    float4 sc = ((const float4*)scale)[d4];
    float4 sh = ((const float4*)shift)[d4];
    float4 y;
    y.x = fmaf(sc.x, a.x, sh.x);
    y.y = fmaf(sc.y, a.y, sh.y);
    y.z = fmaf(sc.z, a.z, sh.z);
    y.w = fmaf(sc.w, a.w, sh.w);
    if (do_relu) {
        y.x = fmaxf(y.x, 0.0f);
        y.y = fmaxf(y.y, 0.0f);
        y.z = fmaxf(y.z, 0.0f);
        y.w = fmaxf(y.w, 0.0f);
    }
    if (do_resid) {
        float4 xv = ((const float4*)xin)[i];
        y.x += xv.x; y.y += xv.y; y.z += xv.z; y.w += xv.w;
    }
    ((float4*)xout)[i] = y;
}

// ------------------------------- launcher -----------------------------------
extern "C" void kernel_launch(void* const* d_in, const int* in_sizes, int n_in,
                              void* d_out, int out_size, void* d_ws, size_t ws_size,
                              hipStream_t stream)
{
    const float* x      = (const float*)d_in[0];
    const int*   eidx   = (const int*)  d_in[1];
    const float* Ws     = (const float*)d_in[2];
    const float* bs     = (const float*)d_in[3];
    const float* gammas = (const float*)d_in[4];
    const float* betas  = (const float*)d_in[5];

    const int N = in_sizes[0] / DFEAT;            // 50000
    const int E = in_sizes[1] / 2;                // 800000
    const int L = in_sizes[2] / (DFEAT * DFEAT);  // 3

    const int* src = eidx;
    const int* dst = eidx + E;

    float* out = (float*)d_out;                   // running x lives here

    // scratch carve-out
    char*  ws  = (char*)d_ws;
    size_t off = 0;
    auto take = [&](size_t bytes) -> float* {
        float* p = (float*)(ws + off);
        off += (bytes + 255) & ~(size_t)255;
        return p;
    };
    float* h     = take((size_t)N * DFEAT * sizeof(float));  // 25.6 MB
    float* agg   = take((size_t)N * DFEAT * sizeof(float));  // 25.6 MB
    float* dis   = take((size_t)N * sizeof(float));          // deg -> deg^-1/2
    float* normw = take((size_t)E * sizeof(float));          // 3.2 MB
    float* sums  = take(DFEAT * sizeof(float));
    float* sumsq = take(DFEAT * sizeof(float));
    float* scale = take(DFEAT * sizeof(float));
    float* shift = take(DFEAT * sizeof(float));

    const dim3 b256(256);
    const int  elem4 = N * (DFEAT / 4);

    // graph normalization (once)
    deg_init <<<dim3((N + 255) / 256), b256, 0, stream>>>(dis, N);
    deg_count<<<dim3((E + 255) / 256), b256, 0, stream>>>(dst, dis, E);
    make_dis <<<dim3((N + 255) / 256), b256, 0, stream>>>(dis, N);
    make_norm<<<dim3((E + 255) / 256), b256, 0, stream>>>(src, dst, dis, normw, E);

    for (int i = 0; i < L; ++i) {
        const float* xin = (i == 0) ? x : out;
        gemm_xw_wmma<<<dim3((N + 127) / 128), b256, 0, stream>>>(
            xin, Ws + (size_t)i * DFEAT * DFEAT, h, N);
        init_agg<<<dim3((elem4 + 255) / 256), b256, 0, stream>>>(
            h, dis, bs + (size_t)i * DFEAT, agg, N);
        scatter_edges<<<dim3((E * 32 + 255) / 256), b256, 0, stream>>>(
            src, dst, normw, h, agg, E);
        zero_stats<<<dim3(1), b256, 0, stream>>>(sums, sumsq);
        col_stats<<<dim3((N + 511) / 512), b256, 0, stream>>>(agg, sums, sumsq, N);
        bn_finalize<<<dim3(1), dim3(DFEAT), 0, stream>>>(
            sums, sumsq, gammas + (size_t)i * DFEAT, betas + (size_t)i * DFEAT,
            scale, shift, N);
        bn_apply<<<dim3((elem4 + 255) / 256), b256, 0, stream>>>(
            agg, scale, shift, out, out, N, (i < L - 1) ? 1 : 0, (i > 0) ? 1 : 0);
    }
    (void)n_in; (void)out_size; (void)ws_size;
}